// FilterLayer_4664334483556
// MI455X (gfx1250) — compile-verified
//
#include <hip/hip_runtime.h>
#include <stdint.h>

// Kernel-prediction filter: y[b,c,h,w] = sum_{k=0..24} xpad[b,c,h+i,w+j] * f[b,k,h,w]
// B=4, C=3, H=W=512, window=5 (pad=2). Memory-bound (AI ~1.2 FLOP/B).
//
// CDNA5 path: stage the haloed x tile into LDS with global_load_async_to_lds_b32
// (ASYNCcnt DMA), drain with s_wait_asynccnt, stream f with non-temporal loads.

#define TILE_W 32
#define TILE_H 8
#define HALO   2
#define WIN    5
#define SH     (TILE_H + 2 * HALO)   // 12
#define SW     (TILE_W + 2 * HALO)   // 36

__global__ __launch_bounds__(256) void kpn_filter_kernel(
    const float* __restrict__ x,   // [B,3,H,W]
    const float* __restrict__ f,   // [B,25,H,W]
    float* __restrict__ y,         // [B,3,H,W]
    int H, int W)
{
    __shared__ float sx[3][SH][SW];  // 5184 B

    const int tx  = threadIdx.x;             // 0..31
    const int ty  = threadIdx.y;             // 0..7
    const int tid = ty * TILE_W + tx;        // 0..255
    const int w0  = blockIdx.x * TILE_W;
    const int h0  = blockIdx.y * TILE_H;
    const int b   = blockIdx.z;
    const int HW  = H * W;

    // ---- Stage x tile (3 channels + halo, zero-padded) into LDS ----
    // In-range elements: async global->LDS DMA (gfx1250, ASYNCcnt).
    // Out-of-range halo: ds_store of 0 (disjoint LDS addresses, no hazard).
    const uint32_t lds_base = (uint32_t)(uintptr_t)(&sx[0][0][0]);
    const uint64_t xb = (uint64_t)(uintptr_t)(x + (size_t)b * 3 * HW);

    const int NELEM = 3 * SH * SW;  // 1296
    for (int idx = tid; idx < NELEM; idx += TILE_W * TILE_H) {
        int c   = idx / (SH * SW);
        int rem = idx - c * (SH * SW);
        int r   = rem / SW;
        int col = rem - r * SW;
        int gh  = h0 - HALO + r;
        int gw  = w0 - HALO + col;
        uint32_t lds_off = lds_base + (uint32_t)idx * 4u;
        if ((unsigned)gh < (unsigned)H && (unsigned)gw < (unsigned)W) {
            uint32_t voff = (uint32_t)((c * HW + gh * W + gw) * 4);
            // GVS mode: mem_addr = SGPR64 + VGPR32; dsaddr = VGPR(lds_off)
            asm volatile("global_load_async_to_lds_b32 %0, %1, %2"
                         :
                         : "v"(lds_off), "v"(voff), "s"(xb)
                         : "memory");
        } else {
            ((float*)&sx[0][0][0])[idx] = 0.0f;
        }
    }

    // Drain this wave's async DMAs, then barrier so all waves' tiles are visible.
    asm volatile("s_wait_asynccnt 0" ::: "memory");
    __syncthreads();

    // ---- Per-pixel weighted sum over the 5x5 window ----
    const int h = h0 + ty;
    const int w = w0 + tx;
    if (h < H && w < W) {
        const float* fp = f + (size_t)b * 25 * HW + (size_t)h * W + w;
        float acc0 = 0.0f, acc1 = 0.0f, acc2 = 0.0f;
#pragma unroll
        for (int i = 0; i < WIN; ++i) {
#pragma unroll
            for (int j = 0; j < WIN; ++j) {
                // f is streamed once: non-temporal, coalesced (wave32 x 4B = 128B line)
                float fv = __builtin_nontemporal_load(fp + (size_t)(i * WIN + j) * HW);
                acc0 = fmaf(sx[0][ty + i][tx + j], fv, acc0);
                acc1 = fmaf(sx[1][ty + i][tx + j], fv, acc1);
                acc2 = fmaf(sx[2][ty + i][tx + j], fv, acc2);
            }
        }
        float* yp = y + (size_t)b * 3 * HW + (size_t)h * W + w;
        __builtin_nontemporal_store(acc0, yp);
        __builtin_nontemporal_store(acc1, yp + HW);
        __builtin_nontemporal_store(acc2, yp + 2 * HW);
    }
}

extern "C" void kernel_launch(void* const* d_in, const int* in_sizes, int n_in,
                              void* d_out, int out_size, void* d_ws, size_t ws_size,
                              hipStream_t stream) {
    const float* x = (const float*)d_in[0];  // [4,3,512,512] f32
    const float* f = (const float*)d_in[1];  // [4,25,512,512] f32
    // d_in[2] = window scalar (5), fixed at compile time.
    float* y = (float*)d_out;                // [4,3,512,512] f32

    const int B = 4, H = 512, W = 512;
    (void)in_sizes; (void)n_in; (void)out_size; (void)d_ws; (void)ws_size;

    dim3 block(TILE_W, TILE_H, 1);
    dim3 grid(W / TILE_W, H / TILE_H, B);
    kpn_filter_kernel<<<grid, block, 0, stream>>>(x, f, y, H, W);
}